// conv_2d_44169443672908
// MI455X (gfx1250) — compile-verified
//
#include <hip/hip_runtime.h>

typedef __attribute__((ext_vector_type(2))) float v2f;
typedef __attribute__((ext_vector_type(4))) float v4f;
typedef __attribute__((ext_vector_type(8))) float v8f;
typedef __attribute__((ext_vector_type(4))) int   v4i;

#ifndef __has_builtin
#define __has_builtin(x) 0
#endif

#if __has_builtin(__builtin_amdgcn_global_load_async_to_lds_b128)
#define USE_ASYNC 1
#else
#define USE_ASYNC 0
#endif

// Builtin parameter types (from the round-2 diagnostic): int4 in AS1 (global /
// __device__) for src, and by symmetry int4 in AS3 (LDS / __shared__) for dst.
typedef __attribute__((address_space(1))) v4i gv4i;
typedef __attribute__((address_space(3))) v4i lv4i;

namespace {
constexpr int Hh = 128, Ww = 128, Cc = 128;
constexpr int Ff = 256, KH = 3, KW = 3;
constexpr int OH = 126, OW = 126;
constexpr int Kdim = KH * KW * Cc;          // 1152
constexpr int Mdim = 16 * OH * OW;          // 254016
constexpr int BM = 64, BN = 128, BK = 32;
constexpr int KC = Kdim / BK;               // 36 k-chunks
constexpr int ROWW = BK + 4;                // 36-word padded LDS row (bank-conflict-free)
constexpr int A_REGION = BM * ROWW;         // 2304 words
constexpr int B_REGION = BN * ROWW;         // 4608 words
constexpr int STAGE_WORDS = A_REGION + B_REGION;  // 6912 words = 27648 B
constexpr int NBLK_M = Mdim / BM;           // 3969 (exact)
constexpr int NBLK_N = Ff / BN;             // 2
constexpr int NUM_BLOCKS = NBLK_M * NBLK_N; // 7938
}

__device__ __forceinline__ void cp16(float* dst_lds, const float* src_glb) {
#if USE_ASYNC
  __builtin_amdgcn_global_load_async_to_lds_b128(
      (gv4i*)src_glb, (lv4i*)dst_lds, /*imm offset*/0, /*cpol*/0);
#else
  *(v4f*)dst_lds = *(const v4f*)src_glb;
#endif
}

__device__ __forceinline__ void wait_async(bool pending6) {
#if USE_ASYNC
  if (pending6) asm volatile("s_wait_asynccnt 0x6" ::: "memory");
  else          asm volatile("s_wait_asynccnt 0x0" ::: "memory");
#else
  (void)pending6;
#endif
}

// Implicit-GEMM 3x3 VALID conv, NHWC fp32, full fp32 precision via
// V_WMMA_F32_16X16X4_F32. Block tile 64Mx128N, K staged in LDS in 32-wide
// double-buffered chunks moved with GLOBAL_LOAD_ASYNC_TO_LDS_B128.
__global__ __launch_bounds__(256) void conv3x3_wmma_lds(
    const float* __restrict__ x,     // (16,128,128,128) NHWC
    const float* __restrict__ w,     // (256,3,3,128) -> flat f*1152 + k
    const float* __restrict__ bias,  // (256,)
    float* __restrict__ out)         // (16,126,126,256)
{
  __shared__ __align__(16) float smem[2][STAGE_WORDS];

  const int tid    = threadIdx.x;
  const int lane   = tid & 31;
  const int waveId = tid >> 5;
  const int laneLo = lane & 15;
  const int laneHi = lane >> 4;              // 0 / 1

  const int mBase = (blockIdx.x >> 1) * BM;
  const int nBase = (blockIdx.x & 1) * BN;

  // ---------------- staging assignment: tid = r*8 + q ----------------
  const int q  = tid & 7;                    // which 16B chunk of a 128B k-row
  const int r0 = tid >> 3;                   // 0..31

  const float* aSrc[2];                      // A rows r0, r0+32
#pragma unroll
  for (int i = 0; i < 2; ++i) {
    const int m     = mBase + r0 + i * 32;
    const int n_img = m / (OH * OW);
    const int rem   = m % (OH * OW);
    const int oh    = rem / OW, ow = rem % OW;
    aSrc[i] = x + (((long)n_img * Hh + oh) * Ww + ow) * Cc + q * 4;
  }
  const float* bSrc[4];                      // B rows r0 + {0,32,64,96}
#pragma unroll
  for (int i = 0; i < 4; ++i)
    bSrc[i] = w + (long)(nBase + r0 + i * 32) * Kdim + q * 4;

  const int aDst0 = r0 * ROWW + q * 4;
  const int aDst1 = (r0 + 32) * ROWW + q * 4;
  int bDst[4];
#pragma unroll
  for (int i = 0; i < 4; ++i)
    bDst[i] = A_REGION + (r0 + i * 32) * ROWW + q * 4;

  // ---------------- compute-fragment LDS offsets ----------------
  const int waveM = waveId & 1;              // 2 M-positions of 32
  const int waveN = waveId >> 1;             // 4 N-positions of 32
  const int kOff  = laneHi * 2;              // f32 A/B striping: lanes>=16 hold K+2,K+3
  const int rowA0 = (waveM * 32 + laneLo) * ROWW + kOff;
  const int rowA1 = rowA0 + 16 * ROWW;
  const int rowB0 = A_REGION + (waveN * 32 + laneLo) * ROWW + kOff;
  const int rowB1 = rowB0 + 16 * ROWW;

  v8f acc[2][2] = {};

  auto stage = [&](int buf, int kc) {
    const int slice = kc >> 2;                 // (kh,kw) slice: 128 k's = 4 chunks
    const int c0    = (kc & 3) << 5;
    const int kh    = slice / 3, kw = slice - kh * 3;
    const int xoff  = (kh * Ww + kw) * Cc + c0;   // offset into x row window
    const int woff  = kc * BK;                    // offset into w's flat K
    float* s = &smem[buf][0];
    cp16(s + aDst0, aSrc[0] + xoff);
    cp16(s + aDst1, aSrc[1] + xoff);
#pragma unroll
    for (int i = 0; i < 4; ++i)
      cp16(s + bDst[i], bSrc[i] + woff);         // 6 async b128 per wave per stage
  };

  stage(0, 0);
  for (int kc = 0; kc < KC; ++kc) {
    const int  cur  = kc & 1;
    const bool more = (kc + 1) < KC;
    if (more) stage(cur ^ 1, kc + 1);
    wait_async(more);          // own stage(cur) done (in-order; 6 newer outstanding)
    __syncthreads();           // all waves' stage(cur) done

    const float* As = &smem[cur][0];
#pragma unroll
    for (int ks = 0; ks < 8; ++ks) {           // 8 K=4 steps per 32-chunk
      v2f a0 = *(const v2f*)(As + rowA0 + ks * 4);
      v2f a1 = *(const v2f*)(As + rowA1 + ks * 4);
      v2f b0 = *(const v2f*)(As + rowB0 + ks * 4);
      v2f b1 = *(const v2f*)(As + rowB1 + ks * 4);
      acc[0][0] = __builtin_amdgcn_wmma_f32_16x16x4_f32(false, a0, false, b0, (short)0, acc[0][0], false, false);
      acc[0][1] = __builtin_amdgcn_wmma_f32_16x16x4_f32(false, a0, false, b1, (short)0, acc[0][1], false, false);
      acc[1][0] = __builtin_amdgcn_wmma_f32_16x16x4_f32(false, a1, false, b0, (short)0, acc[1][0], false, false);
      acc[1][1] = __builtin_amdgcn_wmma_f32_16x16x4_f32(false, a1, false, b1, (short)0, acc[1][1], false, false);
    }
    __syncthreads();           // safe to overwrite buf cur two stages later
  }

  // ---------------- epilogue: f32 C/D layout ----------------
  // VGPR e -> row e (+8 for lanes 16-31), col = lane&15.
  float bv[2];
#pragma unroll
  for (int j = 0; j < 2; ++j)
    bv[j] = bias[nBase + waveN * 32 + j * 16 + laneLo];

#pragma unroll
  for (int i = 0; i < 2; ++i) {
    const int mSub = mBase + waveM * 32 + i * 16 + (laneHi << 3);
#pragma unroll
    for (int e = 0; e < 8; ++e) {
      const int m     = mSub + e;
      const int n_img = m / (OH * OW);
      const int rem   = m % (OH * OW);
      const int oh    = rem / OW, ow = rem % OW;
      const long ob   = (((long)n_img * OH + oh) * OW + ow) * Ff;
#pragma unroll
      for (int j = 0; j < 2; ++j) {
        const int f = nBase + waveN * 32 + j * 16 + laneLo;
        out[ob + f] = acc[i][j][e] + bv[j];
      }
    }
  }
}

extern "C" void kernel_launch(void* const* d_in, const int* in_sizes, int n_in,
                              void* d_out, int out_size, void* d_ws, size_t ws_size,
                              hipStream_t stream) {
  (void)in_sizes; (void)n_in; (void)out_size; (void)d_ws; (void)ws_size;
  const float* x    = (const float*)d_in[0];
  const float* w    = (const float*)d_in[1];
  const float* bias = (const float*)d_in[2];
  float* out        = (float*)d_out;

  dim3 grid(NUM_BLOCKS);   // 7938 blocks
  dim3 block(256);         // 8 wave32 waves (2M x 4N of 32x32 tiles)
  conv3x3_wmma_lds<<<grid, block, 0, stream>>>(x, w, bias, out);
}